// ROIAlign_48404281426054
// MI455X (gfx1250) — compile-verified
//
#include <hip/hip_runtime.h>

// ---------------------------------------------------------------------------
// ROI-Align (aligned=True, scale=0.25, sampling_ratio=2, 7x7) on gfx1250.
//
// Separable factorization per ROI:
//   out[ph,pw,c] = CY(7x28) * G(28x28,c) * CX(7x28)^T
// where G is the gathered corner grid (exactly the same 784 feature positions
// the direct method reads) and CY/CX fold bilinear fractions, the 1/4 pooling
// average (0.5 per axis) and validity masks. Both small GEMMs run on
// V_WMMA_F32_16X16X4_F32 (full fp32 — same precision class as reference).
//
// One block per ROI, 4 waves; each wave owns 4 channel-blocks of 16.
// A-fragments and row offsets are hoisted; B gathers are software-pipelined
// one k-iteration ahead of the WMMA chain to hide L2 gather latency.
// ---------------------------------------------------------------------------

typedef __attribute__((ext_vector_type(2))) float v2f;
typedef __attribute__((ext_vector_type(8))) float v8f;

#define POOL   7
#define GRID   14          // POOL * sampling_ratio
#define NSLOT  28          // GRID * 2 corner slots per axis
#define CHAN   256
#define FH     200
#define FW     272
#define SCALE  0.25f
#define WAVES  4
#define PASSES 4           // 16 channel-blocks / 4 waves

__global__ __launch_bounds__(128) void roi_align_wmma_f32(
    const float* __restrict__ feat,
    const float* __restrict__ rois,
    float* __restrict__ out)
{
    __shared__ int   sRow[NSLOT];
    __shared__ float sWy [NSLOT];
    __shared__ int   sCol[NSLOT];
    __shared__ float sWx [NSLOT];
    __shared__ int   sB;
    // per-wave intermediate T[ph][k][c16] for stage B
    __shared__ float sT[WAVES][POOL][NSLOT][16];

    const int roi  = blockIdx.x;
    const int tid  = threadIdx.x;
    const int lane = tid & 31;
    const int wid  = tid >> 5;

    // ---- per-ROI sampling geometry: threads 0..13 -> y axis, 16..29 -> x ----
    if (tid < 32) {
        const float rx1 = rois[roi * 5 + 1] * SCALE - 0.5f;
        const float ry1 = rois[roi * 5 + 2] * SCALE - 0.5f;
        const float rx2 = rois[roi * 5 + 3] * SCALE - 0.5f;
        const float ry2 = rois[roi * 5 + 4] * SCALE - 0.5f;
        if (tid == 0) sB = (int)rois[roi * 5 + 0];
        const float binW = (rx2 - rx1) / (float)POOL;
        const float binH = (ry2 - ry1) / (float)POOL;
        const int g = tid & 15;
        if (g < GRID) {
            const bool  doY   = (tid < 16);
            const float start = doY ? ry1  : rx1;
            const float bin   = doY ? binH : binW;
            const int   isize = doY ? FH   : FW;
            // sample coordinate: start + floor(g/2)*bin + ((g&1)+0.5)*bin/2
            const float v = start + (float)(g >> 1) * bin
                          + ((float)(g & 1) + 0.5f) * bin * 0.5f;
            const bool  valid = (v > -1.0f) && (v < (float)isize);
            float vc  = fminf(fmaxf(v, 0.0f), (float)isize - 1.0f);
            float flo = floorf(vc);
            int   lo  = (int)flo;
            int   hi  = lo + 1; if (hi > isize - 1) hi = isize - 1;
            float fr  = vc - flo;
            // fold validity mask and 0.5 (per-axis share of the 1/4 average)
            float w0 = valid ? (1.0f - fr) * 0.5f : 0.0f;
            float w1 = valid ? fr * 0.5f          : 0.0f;
            if (doY) { sRow[2*g] = lo; sRow[2*g+1] = hi; sWy[2*g] = w0; sWy[2*g+1] = w1; }
            else     { sCol[2*g] = lo; sCol[2*g+1] = hi; sWx[2*g] = w0; sWx[2*g+1] = w1; }
        }
    }
    __syncthreads();

    const int    b     = sB;
    const int    m     = lane & 15;   // A-row (M) and B/D column (N) for this lane
    const int    half  = lane >> 4;   // K-split: VGPR0 holds K=2*half, VGPR1 K=2*half+1
    const size_t plane = (size_t)FH * FW;
    const float* fbase = feat + (size_t)b * CHAN * plane;

    // ---- hoisted per-ROI A-fragments (block-diagonal CY/CX chunks) and ----
    // ---- per-lane row offsets; branch-free so no predicated LDS loads.  ----
    v2f aY[7], aX[7];
    int rowOff[14];
    #pragma unroll
    for (int jc = 0; jc < 7; ++jc) {
        const float sel = (m == jc) ? 1.0f : 0.0f;   // v_cndmask, then v_mul
        aY[jc].x = sWy[4*jc + 2*half + 0] * sel;
        aY[jc].y = sWy[4*jc + 2*half + 1] * sel;
        aX[jc].x = sWx[4*jc + 2*half + 0] * sel;
        aX[jc].y = sWx[4*jc + 2*half + 1] * sel;
        rowOff[2*jc + 0] = sRow[4*jc + 2*half + 0] * FW;
        rowOff[2*jc + 1] = sRow[4*jc + 2*half + 1] * FW;
    }

    for (int pass = 0; pass < PASSES; ++pass) {
        const int    cblk  = pass * WAVES + wid;
        const int    c     = cblk * 16 + m;             // this lane's channel
        const float* cbase = fbase + (size_t)c * plane;

        // ---- stage A: T[ph,k,c] = sum_j CY[ph,j] * G[j,k,c] ----
        // Software pipeline: gather B for k+1 while WMMA-chaining k.
        v2f bf[7];
        {
            const int col0 = sCol[0];
            #pragma unroll
            for (int jc = 0; jc < 7; ++jc) {
                bf[jc].x = cbase[rowOff[2*jc + 0] + col0];
                bf[jc].y = cbase[rowOff[2*jc + 1] + col0];
            }
        }
        for (int k = 0; k < NSLOT; ++k) {
            const int kn   = (k + 1 < NSLOT) ? k + 1 : k;  // dummy reload on last
            const int coln = sCol[kn];
            v2f bn[7];
            #pragma unroll
            for (int jc = 0; jc < 7; ++jc) {               // prefetch next k
                bn[jc].x = cbase[rowOff[2*jc + 0] + coln];
                bn[jc].y = cbase[rowOff[2*jc + 1] + coln];
            }
            v8f acc = {};
            #pragma unroll
            for (int jc = 0; jc < 7; ++jc)
                acc = __builtin_amdgcn_wmma_f32_16x16x4_f32(
                          false, aY[jc], false, bf[jc], (short)0, acc, false, false);
            if (lane < 16) {
                #pragma unroll
                for (int ph = 0; ph < POOL; ++ph)
                    sT[wid][ph][k][m] = acc[ph];   // D row v = M=v (lanes 0..15)
            }
            #pragma unroll
            for (int jc = 0; jc < 7; ++jc)
                bf[jc] = bn[jc];
        }

        // ---- stage B: out[ph,pw,c] = sum_k CX[pw,k] * T[ph,k,c] ----
        for (int ph = 0; ph < POOL; ++ph) {
            v2f b2[7];
            #pragma unroll
            for (int kc = 0; kc < 7; ++kc) {               // batch LDS reads
                const int k0 = 4*kc + 2*half;
                b2[kc].x = sT[wid][ph][k0    ][m];
                b2[kc].y = sT[wid][ph][k0 + 1][m];
            }
            v8f acc2 = {};
            #pragma unroll
            for (int kc = 0; kc < 7; ++kc)
                acc2 = __builtin_amdgcn_wmma_f32_16x16x4_f32(
                           false, aX[kc], false, b2[kc], (short)0, acc2, false, false);
            if (lane < 16) {
                float* o = out + (((size_t)roi * CHAN + c) * POOL + ph) * POOL;
                #pragma unroll
                for (int pw = 0; pw < POOL; ++pw)
                    o[pw] = acc2[pw];
            }
        }
    }
}

extern "C" void kernel_launch(void* const* d_in, const int* in_sizes, int n_in,
                              void* d_out, int out_size, void* d_ws, size_t ws_size,
                              hipStream_t stream) {
    const float* feat = (const float*)d_in[0];
    const float* rois = (const float*)d_in[1];
    const int n_rois  = in_sizes[1] / 5;
    roi_align_wmma_f32<<<dim3(n_rois), dim3(128), 0, stream>>>(
        feat, rois, (float*)d_out);
}